// MultiBoxLoss_10651518894565
// MI455X (gfx1250) — compile-verified
//
#include <hip/hip_runtime.h>

#define B_ 64
#define P_ 16800
#define T_ 48
#define TG 15
#define THRESH 0.35f
#define NEGPOS 7
#define VAR0 0.1f
#define VAR1 0.2f

typedef float v2f __attribute__((ext_vector_type(2)));
typedef float v8f __attribute__((ext_vector_type(8)));

// ---------------- workspace layout (bytes) ----------------
static constexpr size_t SZ_BP    = (size_t)B_ * P_;
static constexpr size_t OFF_BTOV  = 0;
static constexpr size_t OFF_BTIDX = OFF_BTOV  + SZ_BP * 4;
static constexpr size_t OFF_LOSSC = OFF_BTIDX + SZ_BP * 4;
static constexpr size_t OFF_BEST  = OFF_LOSSC + SZ_BP * 4;          // u64 [B*T]
static constexpr size_t OFF_NPOS  = OFF_BEST  + (size_t)B_ * T_ * 8;
static constexpr size_t OFF_NPOS1 = OFF_NPOS  + B_ * 4;
static constexpr size_t OFF_LL    = OFF_NPOS1 + B_ * 4;
static constexpr size_t OFF_LM    = OFF_LL    + B_ * 4;
static constexpr size_t OFF_PCE   = OFF_LM    + B_ * 4;
static constexpr size_t OFF_NCE   = OFF_PCE   + B_ * 4;

// ---------------- LDS staging (async copy path on gfx1250) ----------------
#if __has_builtin(__builtin_amdgcn_global_load_async_to_lds_b32)
#define HAVE_ASYNC_LDS 1
typedef __attribute__((address_space(1))) int* as1_i32p;
typedef __attribute__((address_space(3))) int* as3_i32p;
#endif

__device__ __forceinline__ void stage_to_lds(const float* __restrict__ g,
                                             float* l, int n) {
  const int tid = threadIdx.x, nt = blockDim.x;
#if defined(HAVE_ASYNC_LDS)
  for (int i = tid; i < n; i += nt) {
    __builtin_amdgcn_global_load_async_to_lds_b32(
        (as1_i32p)(g + i), (as3_i32p)(l + i), 0, 0);
  }
#if __has_builtin(__builtin_amdgcn_s_wait_asynccnt)
  __builtin_amdgcn_s_wait_asynccnt(0);
#else
  asm volatile("s_wait_asynccnt 0" ::: "memory");
#endif
#else
  for (int i = tid; i < n; i += nt) l[i] = g[i];
#endif
  __syncthreads();
}

// ---------------- deterministic reductions ----------------
__device__ __forceinline__ float wave_sum_f(float v) {
  for (int o = 16; o > 0; o >>= 1) v += __shfl_down(v, o);
  return v;
}
__device__ __forceinline__ int wave_sum_i(int v) {
  for (int o = 16; o > 0; o >>= 1) v += __shfl_down(v, o);
  return v;
}
// result valid on thread 0 only
__device__ __forceinline__ float block_sum_f(float v, float* rf) {
  const int lane = threadIdx.x & 31, wid = threadIdx.x >> 5;
  const int nw = (blockDim.x + 31) >> 5;
  v = wave_sum_f(v);
  __syncthreads();
  if (lane == 0) rf[wid] = v;
  __syncthreads();
  float r = 0.f;
  if (threadIdx.x == 0) for (int i = 0; i < nw; ++i) r += rf[i];
  return r;
}
__device__ __forceinline__ int block_sum_i(int v, int* ri) {
  const int lane = threadIdx.x & 31, wid = threadIdx.x >> 5;
  const int nw = (blockDim.x + 31) >> 5;
  v = wave_sum_i(v);
  __syncthreads();
  if (lane == 0) ri[wid] = v;
  __syncthreads();
  int r = 0;
  if (threadIdx.x == 0) for (int i = 0; i < nw; ++i) r += ri[i];
  return r;
}

// ---------------- K0: zero per-(b,t) best-prior keys ----------------
__global__ void k_init(unsigned long long* __restrict__ best_pt) {
  int i = blockIdx.x * blockDim.x + threadIdx.x;
  if (i < B_ * T_) best_pt[i] = 0ull;
}

// ---------------- K1: IoU match ----------------
// per-prior best truth (strict > keeps first index), per-truth best prior via
// packed u64 max: (ov_bits << 32) | (~p)  -> max picks biggest ov, smallest p.
__global__ void k_match(const float* __restrict__ priors,
                        const float* __restrict__ targets,
                        float* __restrict__ bt_ov, int* __restrict__ bt_idx,
                        unsigned long long* __restrict__ best_pt) {
  __shared__ float sT[T_ * TG];
  __shared__ float sAreaT[T_];
  __shared__ unsigned long long sBest[T_];
  const int b = blockIdx.y;
  stage_to_lds(targets + (size_t)b * T_ * TG, sT, T_ * TG);
  const int tid = threadIdx.x;
  if (tid < T_) {
    const float* t = sT + tid * TG;
    sAreaT[tid] = (t[2] - t[0]) * (t[3] - t[1]);
    sBest[tid]  = 0ull;
  }
  __syncthreads();
  const int p = blockIdx.x * blockDim.x + tid;
  if (p < P_) {
    int pn = p + 256; if (pn > P_ - 1) pn = P_ - 1;
    __builtin_prefetch(priors + 4 * pn, 0, 0);
    const float4 pv = *(const float4*)(priors + 4 * p);   // b128
    float x1 = pv.x - 0.5f * pv.z, y1 = pv.y - 0.5f * pv.w;
    float x2 = pv.x + 0.5f * pv.z, y2 = pv.y + 0.5f * pv.w;
    float areaP = (x2 - x1) * (y2 - y1);
    float bestOv = 0.f; int bestT = 0;
    const unsigned long long lo = (unsigned long long)(0xFFFFFFFFu - (unsigned)p);
    for (int t = 0; t < T_; ++t) {
      const float* tr = sT + t * TG;
      float iw = fmaxf(fminf(tr[2], x2) - fmaxf(tr[0], x1), 0.f);
      float ih = fmaxf(fminf(tr[3], y2) - fmaxf(tr[1], y1), 0.f);
      float inter = iw * ih;
      float ov = inter / (sAreaT[t] + areaP - inter);
      if (ov > bestOv) { bestOv = ov; bestT = t; }
      unsigned long long key = ((unsigned long long)__float_as_uint(ov) << 32) | lo;
      atomicMax(&sBest[t], key);
    }
    bt_ov [(size_t)b * P_ + p] = bestOv;
    bt_idx[(size_t)b * P_ + p] = bestT;
  }
  __syncthreads();
  if (tid < T_) atomicMax(&best_pt[b * T_ + tid], sBest[tid]);
}

// ---------------- K2: scatter override (sequential per batch: last wins) ----
__global__ void k_override(const unsigned long long* __restrict__ best_pt,
                           float* __restrict__ bt_ov, int* __restrict__ bt_idx) {
  int b = threadIdx.x;
  if (b >= B_) return;
  for (int t = 0; t < T_; ++t) {
    unsigned idx = 0xFFFFFFFFu - (unsigned)(best_pt[b * T_ + t] & 0xFFFFFFFFull);
    bt_ov [(size_t)b * P_ + idx] = 2.0f;
    bt_idx[(size_t)b * P_ + idx] = t;
  }
}

// ---------------- K3: per-(b,p) losses, loss_c, per-batch partials ----------
__global__ void k_loss(const float* __restrict__ loc,
                       const float* __restrict__ conf,
                       const float* __restrict__ lmd,
                       const float* __restrict__ priors,
                       const float* __restrict__ targets,
                       const float* __restrict__ bt_ov,
                       const int* __restrict__ bt_idx,
                       float* __restrict__ loss_c,
                       int* __restrict__ nposB, int* __restrict__ npos1B,
                       float* __restrict__ llB, float* __restrict__ lmB,
                       float* __restrict__ pceB) {
  __shared__ float sT[T_ * TG];
  __shared__ float rf[8];
  __shared__ int   ri[8];
  const int b = blockIdx.x;
  stage_to_lds(targets + (size_t)b * T_ * TG, sT, T_ * TG);

  float accL = 0.f, accLm = 0.f, accP = 0.f;
  int cPos = 0, cPos1 = 0;

  for (int p = threadIdx.x; p < P_; p += blockDim.x) {
    const size_t gi = (size_t)b * P_ + p;
    {
      int pn = p + blockDim.x; if (pn > P_ - 1) pn = P_ - 1;
      const size_t gn = (size_t)b * P_ + pn;
      __builtin_prefetch(lmd  + gn * 10, 0, 0);
      __builtin_prefetch(loc  + gn * 4,  0, 0);
      __builtin_prefetch(conf + gn * 2,  0, 0);
    }
    const int   j  = bt_idx[gi];
    const float ov = bt_ov[gi];
    const float* tr = sT + j * TG;
    const int  conf_t = (ov < THRESH) ? 0 : (int)tr[14];
    const bool pos  = conf_t != 0;
    const bool pos1 = conf_t > 0;

    const float4 pv = *(const float4*)(priors + 4 * p);   // b128
    const float px = pv.x, py = pv.y, pw = pv.z, ph = pv.w;

    // smooth-L1 landmark loss (5x float2 = b64 loads, 8B-aligned: stride 40B)
    float s = 0.f;
    const float ivx = 1.f / (VAR0 * pw), ivy = 1.f / (VAR0 * ph);
    const float2* lmp = (const float2*)(lmd + gi * 10);
    for (int q = 0; q < 5; ++q) {
      const float2 lq = lmp[q];
      float gx = (tr[4 + 2*q]     - px) * ivx;
      float gy = (tr[4 + 2*q + 1] - py) * ivy;
      float dx = lq.x - gx;
      float dy = lq.y - gy;
      float ax = fabsf(dx), ay = fabsf(dy);
      s += (ax < 1.f) ? 0.5f * dx * dx : ax - 0.5f;
      s += (ay < 1.f) ? 0.5f * dy * dy : ay - 0.5f;
    }
    if (pos1) { accLm += s; ++cPos1; }

    // decode + GIoU (1x b128 load)
    const float4 lv = *(const float4*)(loc + gi * 4);
    float cx = px + lv.x * VAR0 * pw, cy = py + lv.y * VAR0 * ph;
    float w  = pw * expf(lv.z * VAR1), h = ph * expf(lv.w * VAR1);
    float bx1 = cx - 0.5f*w, by1 = cy - 0.5f*h, bx2 = cx + 0.5f*w, by2 = cy + 0.5f*h;
    float tx1 = tr[0], ty1 = tr[1], tx2 = tr[2], ty2 = tr[3];
    float a1 = (bx2 - bx1) * (by2 - by1);
    float a2 = (tx2 - tx1) * (ty2 - ty1);
    float iw = fmaxf(fminf(bx2, tx2) - fmaxf(bx1, tx1), 0.f);
    float ih = fmaxf(fminf(by2, ty2) - fmaxf(by1, ty1), 0.f);
    float inter = iw * ih;
    float ow = fmaxf(fmaxf(bx2, tx2) - fminf(bx1, tx1), 0.f);
    float oh = fmaxf(fmaxf(by2, ty2) - fminf(by1, ty1), 0.f);
    float clos = ow * oh;
    float uni  = a1 + a2 - inter;
    float giou = fminf(fmaxf(inter / uni - (clos - uni) / clos, -1.f), 1.f);
    if (pos) { accL += 1.f - giou; ++cPos; }

    // CE pieces (1x b64 load)
    const float2 cvp = *(const float2*)(conf + gi * 2);
    float c0 = cvp.x, cv1 = cvp.y;
    float mx  = fmaxf(c0, cv1);
    float lse = mx + logf(expf(c0 - mx) + expf(cv1 - mx));
    float ce  = lse - (conf_t ? cv1 : c0);
    loss_c[gi] = pos ? 0.f : ce;
    if (pos) accP += ce;
  }

  float tl  = block_sum_f(accL,  rf); __syncthreads();
  float tlm = block_sum_f(accLm, rf); __syncthreads();
  float tp  = block_sum_f(accP,  rf); __syncthreads();
  int   tc  = block_sum_i(cPos,  ri); __syncthreads();
  int   tc1 = block_sum_i(cPos1, ri);
  if (threadIdx.x == 0) {
    llB[b] = tl; lmB[b] = tlm; pceB[b] = tp;
    nposB[b] = tc; npos1B[b] = tc1;
  }
}

// ---------------- K4: per-batch radix top-k select + neg CE sum ------------
__global__ void k_select(const float* __restrict__ loss_c,
                         const int* __restrict__ nposB,
                         float* __restrict__ nceB) {
  __shared__ unsigned hist[256];
  __shared__ unsigned sPrefix;
  __shared__ int      sRem;
  __shared__ float    rf[8];
  const int b = blockIdx.x;
  const float4* lc4 = (const float4*)(loss_c + (size_t)b * P_);
  constexpr int N4 = P_ / 4;                       // 16800 % 4 == 0
  int k = nposB[b] * NEGPOS;
  if (k > P_ - 1) k = P_ - 1;
  if (k <= 0) { if (threadIdx.x == 0) nceB[b] = 0.f; return; }
  if (threadIdx.x == 0) { sPrefix = 0u; sRem = k; }
  __syncthreads();

  for (int pass = 0; pass < 4; ++pass) {
    const int shift = 24 - 8 * pass;
    for (int i = threadIdx.x; i < 256; i += blockDim.x) hist[i] = 0u;
    __syncthreads();
    const unsigned prefix = sPrefix;
    const unsigned hiMask = (shift == 24) ? 0u : (0xFFFFFFFFu << (shift + 8));
    for (int p = threadIdx.x; p < N4; p += blockDim.x) {
      const float4 v = lc4[p];                     // b128 load
      unsigned key;
      key = __float_as_uint(v.x);
      if (((key ^ prefix) & hiMask) == 0u) atomicAdd(&hist[(key >> shift) & 0xFF], 1u);
      key = __float_as_uint(v.y);
      if (((key ^ prefix) & hiMask) == 0u) atomicAdd(&hist[(key >> shift) & 0xFF], 1u);
      key = __float_as_uint(v.z);
      if (((key ^ prefix) & hiMask) == 0u) atomicAdd(&hist[(key >> shift) & 0xFF], 1u);
      key = __float_as_uint(v.w);
      if (((key ^ prefix) & hiMask) == 0u) atomicAdd(&hist[(key >> shift) & 0xFF], 1u);
    }
    __syncthreads();
    if (threadIdx.x == 0) {
      int rem = sRem; unsigned cum = 0;
      for (int v = 255; v >= 0; --v) {
        unsigned c = hist[v];
        if (cum + c >= (unsigned)rem) {
          sPrefix = prefix | ((unsigned)v << shift);
          sRem = rem - (int)cum;
          break;
        }
        cum += c;
      }
    }
    __syncthreads();
  }

  const unsigned thr = sPrefix;
  const int krem = sRem;  // ties at thr to include (all share value thr)
  float acc = 0.f;
  for (int p = threadIdx.x; p < N4; p += blockDim.x) {
    const float4 v = lc4[p];
    if (__float_as_uint(v.x) > thr) acc += v.x;
    if (__float_as_uint(v.y) > thr) acc += v.y;
    if (__float_as_uint(v.z) > thr) acc += v.z;
    if (__float_as_uint(v.w) > thr) acc += v.w;
  }
  float tot = block_sum_f(acc, rf);
  if (threadIdx.x == 0) nceB[b] = tot + (float)krem * __uint_as_float(thr);
}

// ---------------- K5: final 64-way reduction (WMMA) + outputs --------------
__global__ void k_final(const float* __restrict__ llB, const float* __restrict__ lmB,
                        const float* __restrict__ pceB, const float* __restrict__ nceB,
                        const int* __restrict__ nposB, const int* __restrict__ npos1B,
                        float* __restrict__ out) {
  const int lane = threadIdx.x;  // exactly 32 threads, EXEC all-ones
  float bl0 = llB[lane],  bl1 = llB[lane + 32];
  float bm0 = lmB[lane],  bm1 = lmB[lane + 32];
  float bc0 = pceB[lane] + nceB[lane];
  float bc1 = pceB[lane + 32] + nceB[lane + 32];

  float sl, sm, sc;
#if __has_builtin(__builtin_amdgcn_wmma_f32_16x16x4_f32)
  // D = ones(16x4) * B(4x16) + 0  =>  D[m][n] = column-sum(n); every row equal.
  // Sum of all 64 staged values = sum over the 16 columns (lanes 0..15 of d[0]).
  v2f ones; ones.x = 1.f; ones.y = 1.f;
  v2f vb;
  v8f cz = {0.f, 0.f, 0.f, 0.f, 0.f, 0.f, 0.f, 0.f};
  v8f d;
  vb.x = bl0; vb.y = bl1;
  d = __builtin_amdgcn_wmma_f32_16x16x4_f32(false, ones, false, vb, (short)0, cz, false, false);
  sl = d[0];
  vb.x = bm0; vb.y = bm1;
  d = __builtin_amdgcn_wmma_f32_16x16x4_f32(false, ones, false, vb, (short)0, cz, false, false);
  sm = d[0];
  vb.x = bc0; vb.y = bc1;
  d = __builtin_amdgcn_wmma_f32_16x16x4_f32(false, ones, false, vb, (short)0, cz, false, false);
  sc = d[0];
  // sum the 16 column values held in each 16-lane half (halves are duplicates)
  for (int o = 8; o > 0; o >>= 1) {
    sl += __shfl_xor(sl, o);
    sm += __shfl_xor(sm, o);
    sc += __shfl_xor(sc, o);
  }
#else
  sl = bl0 + bl1; sm = bm0 + bm1; sc = bc0 + bc1;
  for (int o = 16; o > 0; o >>= 1) {
    sl += __shfl_xor(sl, o);
    sm += __shfl_xor(sm, o);
    sc += __shfl_xor(sc, o);
  }
#endif
  int np = nposB[lane] + nposB[lane + 32];
  int n1 = npos1B[lane] + npos1B[lane + 32];
  for (int o = 16; o > 0; o >>= 1) {
    np += __shfl_xor(np, o);
    n1 += __shfl_xor(n1, o);
  }
  if (lane == 0) {
    float N  = fmaxf((float)np, 1.f);
    float N1 = fmaxf((float)n1, 1.f);
    out[0] = sl / N;   // loss_l / N
    out[1] = sc / N;   // loss_conf / N
    out[2] = sm / N1;  // loss_landm / N1
  }
}

// ---------------- host launcher ----------------
extern "C" void kernel_launch(void* const* d_in, const int* in_sizes, int n_in,
                              void* d_out, int out_size, void* d_ws, size_t ws_size,
                              hipStream_t stream) {
  (void)in_sizes; (void)n_in; (void)out_size; (void)ws_size;
  const float* loc     = (const float*)d_in[0];
  const float* conf    = (const float*)d_in[1];
  const float* lmd     = (const float*)d_in[2];
  const float* priors  = (const float*)d_in[3];
  const float* targets = (const float*)d_in[4];

  char* ws = (char*)d_ws;
  float* bt_ov  = (float*)(ws + OFF_BTOV);
  int*   bt_idx = (int*)  (ws + OFF_BTIDX);
  float* loss_c = (float*)(ws + OFF_LOSSC);
  unsigned long long* best_pt = (unsigned long long*)(ws + OFF_BEST);
  int*   nposB  = (int*)  (ws + OFF_NPOS);
  int*   npos1B = (int*)  (ws + OFF_NPOS1);
  float* llB    = (float*)(ws + OFF_LL);
  float* lmB    = (float*)(ws + OFF_LM);
  float* pceB   = (float*)(ws + OFF_PCE);
  float* nceB   = (float*)(ws + OFF_NCE);
  float* out    = (float*)d_out;

  k_init<<<dim3((B_ * T_ + 255) / 256), dim3(256), 0, stream>>>(best_pt);
  k_match<<<dim3((P_ + 255) / 256, B_), dim3(256), 0, stream>>>(
      priors, targets, bt_ov, bt_idx, best_pt);
  k_override<<<dim3(1), dim3(64), 0, stream>>>(best_pt, bt_ov, bt_idx);
  k_loss<<<dim3(B_), dim3(256), 0, stream>>>(
      loc, conf, lmd, priors, targets, bt_ov, bt_idx,
      loss_c, nposB, npos1B, llB, lmB, pceB);
  k_select<<<dim3(B_), dim3(256), 0, stream>>>(loss_c, nposB, nceB);
  k_final<<<dim3(1), dim3(32), 0, stream>>>(llB, lmB, pceB, nceB, nposB, npos1B, out);
}